// pinModel_88905823027216
// MI455X (gfx1250) — compile-verified
//
#include <hip/hip_runtime.h>

// CDNA5 (gfx1250) fused kernels for the masked-Kronecker polynomial network.
// wave32; V_WMMA_F32_16X16X4_F32 for all dense 16x16 @ 16xN f32 GEMMs.
//
// Pre-pass bakes the order-2/3 column masks into q-major copies of V2/V3 in
// d_ws, so the WMMA hot loop has zero selects / zero exec predication and
// fully coalesced B-fragment loads.

typedef float v2f __attribute__((ext_vector_type(2)));
typedef float v8f __attribute__((ext_vector_type(8)));

__device__ __forceinline__ v8f wmma_f32_16x16x4(v2f a, v2f b, v8f c) {
    // 8 args: (neg_a, A, neg_b, B, c_mod, C, reuse_a, reuse_b)
    return __builtin_amdgcn_wmma_f32_16x16x4_f32(false, a, false, b, (short)0, c,
                                                 false, false);
}

// ---- pre-pass: mask + transpose V3 (4096x8 -> 8x4096) and V2 (256x8 -> 8x256)
__global__ __launch_bounds__(256)
void premask_kernel(const float* __restrict__ V3, const float* __restrict__ V2,
                    float* __restrict__ V3m, float* __restrict__ V2m)
{
    const int idx = blockIdx.x * 256 + threadIdx.x;   // 128 blocks x 256
    if (idx < 8 * 4096) {
        const int q = idx >> 12, c = idx & 4095;
        const int n = c & 15, l = (c >> 4) & 15, j = c >> 8;
        const float v = V3[c * 8 + q];
        V3m[idx] = (n <= l && l <= j) ? v : 0.f;
    }
    if (idx < 8 * 256) {
        const int q = idx >> 8, c = idx & 255;
        const int l = c & 15, j = c >> 4;
        const float v = V2[c * 8 + q];
        V2m[idx] = (l <= j) ? v : 0.f;
    }
}

__global__ __launch_bounds__(256)
void fused_pin_kernel(const float* __restrict__ x,    const float* __restrict__ km1,
                      const float* __restrict__ b1,   const float* __restrict__ U2,
                      const float* __restrict__ V2m,  const float* __restrict__ core2,
                      const float* __restrict__ U3,   const float* __restrict__ V3m,
                      const float* __restrict__ core3,const float* __restrict__ ln_g,
                      const float* __restrict__ ln_b, const float* __restrict__ fc_w,
                      const float* __restrict__ fc_b, float* __restrict__ out)
{
    __shared__ float xs[256];        // x_b, row-major 16x16
    __shared__ float bufA[16 * 256]; // mode-product staging
    __shared__ float bufB[16 * 256];
    __shared__ float o_s[256];       // per-batch feature accumulator
    __shared__ float T2s[64];        // T2[p][q]
    __shared__ float T3s[64];        // T3[p][q]
    __shared__ float red[256];       // layernorm reductions

    const int b    = blockIdx.x;                              // one WG per batch
    const int t    = threadIdx.x;                             // 8 wave32s
    const int lane = t & 31;
    const int wave = __builtin_amdgcn_readfirstlane(t >> 5);  // provably uniform
    const int m16  = lane & 15;      // M row (A frag) / N col (B,C frags)
    const int g    = lane >> 4;      // lane group per ISA VGPR striping

    xs[t] = x[b * 256 + t];
    if (t < 64) { T2s[t] = 0.f; T3s[t] = 0.f; }
    __syncthreads();

    // A fragments of xs: shared by every G1/G2/G3 stage and all q — load once.
    v2f axs[4];
#pragma unroll
    for (int kb = 0; kb < 4; ++kb) {
        axs[kb].x = xs[m16 * 16 + kb * 4 + 2 * g];
        axs[kb].y = xs[m16 * 16 + kb * 4 + 2 * g + 1];
    }

    // ---------------- order 1: o = vec(x) @ km1 + b1 (mask == ones) --------
    {
        float acc = b1[t];
        for (int r = 0; r < 256; ++r)            // xs[r] broadcast, km1 coalesced
            acc = fmaf(xs[r], km1[r * 256 + t], acc);
        o_s[t] = acc;                             // private to thread t until later
    }

    // ---------------- order 3: T3 = U3'^T (x (x) x (x) x) V3' --------------
    // per rank column q: three Tucker mode products by x (16x16 @ 16x256 WMMA
    // GEMMs), then a 4096-long masked contraction with U3.
    for (int q = 0; q < 8; ++q) {
        const float* Vq = V3m + (q << 12);        // masked, q-major: Vq[j*256+(l,n)]
        // G1: bufA[a][(l,n)] = sum_j xs[a][j] * Vq[j][(l,n)]
#pragma unroll
        for (int tt = 0; tt < 2; ++tt) {
            const int tile = wave + 8 * tt;       // uniform
            const int col = tile * 16 + m16;      // (l,n) composite
            v2f bf[4];
#pragma unroll
            for (int kb = 0; kb < 4; ++kb) {      // clean coalesced loads
                const int j0 = kb * 4 + 2 * g;
                bf[kb].x = Vq[j0 * 256 + col];
                bf[kb].y = Vq[(j0 + 1) * 256 + col];
            }
            v8f acc = {0.f, 0.f, 0.f, 0.f, 0.f, 0.f, 0.f, 0.f};
#pragma unroll
            for (int kb = 0; kb < 4; ++kb) acc = wmma_f32_16x16x4(axs[kb], bf[kb], acc);
#pragma unroll
            for (int v = 0; v < 8; ++v) bufA[(v + 8 * g) * 256 + col] = acc[v];
        }
        __syncthreads();

        // G2: bufB[c][(a,n)] = sum_l xs[c][l] * bufA[a][(l,n)]
#pragma unroll
        for (int tt = 0; tt < 2; ++tt) {
            const int tile = wave + 8 * tt;
            const int col = tile * 16 + m16;      // (a,n)
            const int ai = col >> 4, n = col & 15;
            v2f bf[4];
#pragma unroll
            for (int kb = 0; kb < 4; ++kb) {
                const int k0 = kb * 4 + 2 * g;
                bf[kb].x = bufA[ai * 256 + k0 * 16 + n];
                bf[kb].y = bufA[ai * 256 + (k0 + 1) * 16 + n];
            }
            v8f acc = {0.f, 0.f, 0.f, 0.f, 0.f, 0.f, 0.f, 0.f};
#pragma unroll
            for (int kb = 0; kb < 4; ++kb) acc = wmma_f32_16x16x4(axs[kb], bf[kb], acc);
#pragma unroll
            for (int v = 0; v < 8; ++v) bufB[(v + 8 * g) * 256 + col] = acc[v];
        }
        __syncthreads();

        // G3: bufA[e][(a,c)] = sum_n xs[e][n] * bufB[c][(a,n)]
#pragma unroll
        for (int tt = 0; tt < 2; ++tt) {
            const int tile = wave + 8 * tt;
            const int col = tile * 16 + m16;      // (a,c)
            const int ai = col >> 4, c2 = col & 15;
            v2f bf[4];
#pragma unroll
            for (int kb = 0; kb < 4; ++kb) {
                const int k0 = kb * 4 + 2 * g;
                bf[kb].x = bufB[c2 * 256 + ai * 16 + k0];
                bf[kb].y = bufB[c2 * 256 + ai * 16 + k0 + 1];
            }
            v8f acc = {0.f, 0.f, 0.f, 0.f, 0.f, 0.f, 0.f, 0.f};
#pragma unroll
            for (int kb = 0; kb < 4; ++kb) acc = wmma_f32_16x16x4(axs[kb], bf[kb], acc);
#pragma unroll
            for (int v = 0; v < 8; ++v) bufA[(v + 8 * g) * 256 + col] = acc[v];
        }
        __syncthreads();

        // contraction: T3[p][q] += sum_r mrow3(r) * U3[r][p] * Y3(r)
        {
            const int p = t & 7, chunk = t >> 3;  // 32 chunks x 128 rows
            float partial = 0.f;
            for (int rr = 0; rr < 128; ++rr) {
                const int r = chunk * 128 + rr;
                const int e = r & 15, c2 = (r >> 4) & 15, ai = r >> 8;
                const float y3 = bufA[e * 256 + ai * 16 + c2];        // unconditional
                const float w  = (e <= c2 && c2 <= ai) ? y3 : 0.f;    // cndmask
                partial = fmaf(U3[r * 8 + p], w, partial);
            }
            atomicAdd(&T3s[p * 8 + q], partial);  // ds_add_f32
        }
        __syncthreads();
    }

    // ---------------- order 2: T2 = U2'^T (x (x) x) V2' ---------------------
    for (int q = 0; q < 8; ++q) {
        const float* Vq = V2m + (q << 8);         // masked, q-major: Vq[j*16+l]
        if (wave == 0) {                          // Z = xs @ Vq, 16x16
            const int col = m16;                  // l
            v2f bf[4];
#pragma unroll
            for (int kb = 0; kb < 4; ++kb) {
                const int j0 = kb * 4 + 2 * g;
                bf[kb].x = Vq[j0 * 16 + col];
                bf[kb].y = Vq[(j0 + 1) * 16 + col];
            }
            v8f acc = {0.f, 0.f, 0.f, 0.f, 0.f, 0.f, 0.f, 0.f};
#pragma unroll
            for (int kb = 0; kb < 4; ++kb) acc = wmma_f32_16x16x4(axs[kb], bf[kb], acc);
#pragma unroll
            for (int v = 0; v < 8; ++v) bufB[(v + 8 * g) * 256 + col] = acc[v];
        }
        __syncthreads();
        if (wave == 0) {                          // Y = Z @ xs^T, 16x16
            v2f az[4], bf[4];
#pragma unroll
            for (int kb = 0; kb < 4; ++kb) {
                const int k0 = kb * 4 + 2 * g;    // l
                az[kb].x = bufB[m16 * 256 + k0];
                az[kb].y = bufB[m16 * 256 + k0 + 1];
                bf[kb].x = xs[m16 * 16 + k0];     // B[l][k] = xs[k][l], col k = m16
                bf[kb].y = xs[m16 * 16 + k0 + 1];
            }
            v8f acc = {0.f, 0.f, 0.f, 0.f, 0.f, 0.f, 0.f, 0.f};
#pragma unroll
            for (int kb = 0; kb < 4; ++kb) acc = wmma_f32_16x16x4(az[kb], bf[kb], acc);
#pragma unroll
            for (int v = 0; v < 8; ++v) bufA[(v + 8 * g) * 256 + m16] = acc[v];
        }
        __syncthreads();
        {   // T2[p][q] += sum_r mrow2(r) * U2[r][p] * Y[i][k]
            const int p = t & 7, chunk = t >> 3;  // 32 chunks x 8 rows
            float partial = 0.f;
            for (int rr = 0; rr < 8; ++rr) {
                const int r = chunk * 8 + rr;
                const int k = r & 15, i = r >> 4;
                const float yv = bufA[i * 256 + k];
                const float w  = (k <= i) ? yv : 0.f;
                partial = fmaf(U2[r * 8 + p], w, partial);
            }
            atomicAdd(&T2s[p * 8 + q], partial);
        }
        __syncthreads();
    }

    // ---------------- o += T2*core2 + T3*core3 ------------------------------
    {
        float acc = o_s[t];
        for (int pq = 0; pq < 64; ++pq) {
            acc = fmaf(T2s[pq], core2[pq * 256 + t], acc);
            acc = fmaf(T3s[pq], core3[pq * 256 + t], acc);
        }
        o_s[t] = acc;
        red[t] = acc;
    }
    __syncthreads();

    // ---------------- layernorm over 256 ------------------------------------
    for (int s = 128; s > 0; s >>= 1) {
        if (t < s) red[t] += red[t + s];
        __syncthreads();
    }
    const float mu = red[0] * (1.f / 256.f);
    __syncthreads();
    const float d = o_s[t] - mu;
    red[t] = d * d;
    __syncthreads();
    for (int s = 128; s > 0; s >>= 1) {
        if (t < s) red[t] += red[t + s];
        __syncthreads();
    }
    const float var = red[0] * (1.f / 256.f);
    const float y = d * rsqrtf(var + 1e-5f) * ln_g[t] + ln_b[t];
    __syncthreads();
    o_s[t] = y;
    __syncthreads();

    // ---------------- FC 256 -> 128 ------------------------------------------
    if (t < 128) {
        float acc = fc_b[t];
        for (int dd = 0; dd < 256; ++dd)
            acc = fmaf(o_s[dd], fc_w[t * 256 + dd], acc);
        out[b * 128 + t] = acc;
    }
}

extern "C" void kernel_launch(void* const* d_in, const int* in_sizes, int n_in,
                              void* d_out, int out_size, void* d_ws, size_t ws_size,
                              hipStream_t stream) {
    (void)in_sizes; (void)n_in; (void)out_size; (void)ws_size;
    const float* x     = (const float*)d_in[0];
    const float* km1   = (const float*)d_in[1];
    const float* b1    = (const float*)d_in[2];
    const float* U2    = (const float*)d_in[3];
    const float* V2    = (const float*)d_in[4];
    const float* core2 = (const float*)d_in[5];
    const float* U3    = (const float*)d_in[6];
    const float* V3    = (const float*)d_in[7];
    const float* core3 = (const float*)d_in[8];
    const float* ln_g  = (const float*)d_in[9];
    const float* ln_b  = (const float*)d_in[10];
    const float* fc_w  = (const float*)d_in[11];
    const float* fc_b  = (const float*)d_in[12];
    float* out = (float*)d_out;

    // d_ws layout: [0, 128KB) masked q-major V3, [128KB, 136KB) masked V2.
    float* V3m = (float*)d_ws;
    float* V2m = V3m + 8 * 4096;

    premask_kernel<<<dim3(128), dim3(256), 0, stream>>>(V3, V2, V3m, V2m);
    fused_pin_kernel<<<dim3(8), dim3(256), 0, stream>>>(
        x, km1, b1, U2, V2m, core2, U3, V3m, core3, ln_g, ln_b, fc_w, fc_b, out);
}